// YOLOLoss_6339371729724
// MI455X (gfx1250) — compile-verified
//
#include <hip/hip_runtime.h>
#include <math.h>

#define NUM_CLASSES 80
#define APS 3
#define G0s 52
#define G1s 26
#define G2s 13
#define BATCH 64
#define NBOX 32
#define NB (BATCH*NBOX)

#define R0 (BATCH*G0s*G0s*APS)            /* 519168 */
#define R1 (BATCH*G1s*G1s*APS)            /* 129792 */
#define R2 (BATCH*G2s*G2s*APS)            /*  32448 */
#define TOTROWS (R0+R1+R2)                /* 681408 */
#define ROWLEN (5+NUM_CLASSES)            /* 85 */

#define TILE 2048                          /* rows staged per block (tile_dim1, 16-bit OK) */
#define NBLK0 ((R0 + TILE - 1)/TILE)       /* 254 */
#define NBLK1 ((R1 + TILE - 1)/TILE)       /*  64 */
#define NBLK2 ((R2 + TILE - 1)/TILE)       /*  16 */
#define NBLK  (NBLK0 + NBLK1 + NBLK2)      /* 334 */

typedef __attribute__((ext_vector_type(2))) float        v2f;
typedef __attribute__((ext_vector_type(8))) float        v8f;
typedef __attribute__((ext_vector_type(4))) unsigned int v4u;
typedef __attribute__((ext_vector_type(4))) int          v4i;
typedef __attribute__((ext_vector_type(8))) int          v8i;

__device__ __forceinline__ float bce_logits(float x, float z) {
    return fmaxf(x, 0.0f) - x * z + log1pf(expf(-fabsf(x)));
}

// ---- wave32 reduction on the matrix pipe (V_WMMA_F32_16X16X4_F32, B = ones) ----
__device__ __forceinline__ void wave_wmma_reduce_atomic(float partial, float* dst) {
    v2f a; a[0] = partial; a[1] = 0.0f;
    v2f b; b[0] = 1.0f;    b[1] = 1.0f;
    v8f c = {};
    c = __builtin_amdgcn_wmma_f32_16x16x4_f32(false, a, false, b, (short)0, c, false, false);
    unsigned lane = threadIdx.x & 31u;
    if (lane == 0u || lane == 16u) {
        float s = c[0]+c[1]+c[2]+c[3]+c[4]+c[5]+c[6]+c[7];
        atomicAdd(dst, s);
    }
}

// ---- Tensor Data Mover: strided gather of the conf channel into LDS ------------
#if __has_builtin(__builtin_amdgcn_tensor_load_to_lds) && __has_builtin(__builtin_amdgcn_s_wait_tensorcnt)
#define HAVE_TDM 1
__device__ __forceinline__ void tdm_gather_conf(const float* gsrc, unsigned ldsByteAddr,
                                                unsigned rowsRemaining) {
    // D# group0 (ISA 8.3): [1:0]=count=1, [63:32]=lds_addr,
    // [120:64]=global byte addr, [127:126]=type=2.
    unsigned long long ga = (unsigned long long)(uintptr_t)gsrc;
    v4u g0;
    g0[0] = 1u;
    g0[1] = ldsByteAddr;
    g0[2] = (unsigned)(ga & 0xffffffffull);
    g0[3] = (unsigned)((ga >> 32) & 0x01ffffffull) | (2u << 30);
    // D# group1 (ISA 8.4): data_size=2 (4B) @bits17:16; tensor_dim0=85 @79:48;
    // tensor_dim1=rowsRemaining @111:80; tile_dim0=1 @127:112; tile_dim1=TILE @143:128;
    // tile_dim2=0; tensor_dim0_stride=85 @207:160; tensor_dim1_stride=0.
    v8i g1;
    g1[0] = (int)(2u << 16);
    g1[1] = (int)((85u & 0xffffu) << 16);                      // tensor_dim0 lo16
    g1[2] = (int)(((85u >> 16) & 0xffffu) |                    // tensor_dim0 hi16 (=0)
                  ((rowsRemaining & 0xffffu) << 16));          // tensor_dim1 lo16
    g1[3] = (int)(((rowsRemaining >> 16) & 0xffffu) |          // tensor_dim1 hi16
                  (1u << 16));                                 // tile_dim0 = 1
    g1[4] = (int)((unsigned)TILE & 0xffffu);                   // tile_dim1 = TILE, tile_dim2 = 0
    g1[5] = 85;                                                // tensor_dim0_stride lo32
    g1[6] = 0;                                                 // stride0 hi16 | stride1 lo16
    g1[7] = 0;                                                 // tensor_dim1_stride hi32
    v4i z4 = {0,0,0,0};
#if defined(__clang_major__) && (__clang_major__ >= 23)
    v8i z8 = {0,0,0,0,0,0,0,0};
    __builtin_amdgcn_tensor_load_to_lds(g0, g1, z4, z4, z8, 0);
#else
    __builtin_amdgcn_tensor_load_to_lds(g0, g1, z4, z4, 0);
#endif
    __builtin_amdgcn_s_wait_tensorcnt((short)0);
}
#endif

// ---------------- Kernel 0: init workspace (every call; graph-replay safe) ------
__global__ void k_init(int* __restrict__ winner, float* __restrict__ acc) {
    int i = blockIdx.x * blockDim.x + threadIdx.x;
    if (i < 4) acc[i] = 0.0f;
    int stride = gridDim.x * blockDim.x;
    for (int r = i; r < TOTROWS; r += stride) winner[r] = -1;
}

// ---------------- Kernel 1: per-box targets + deterministic winner election -----
__global__ void k_assign(const float* __restrict__ bboxes,
                         const int*   __restrict__ labels,
                         const float* __restrict__ anchors,
                         int* __restrict__ winner,
                         float4* __restrict__ tvals,
                         int* __restrict__ labs) {
    int t = blockIdx.x * blockDim.x + threadIdx.x;
    if (t >= NB) return;

    float x = bboxes[t*4+0], y = bboxes[t*4+1];
    float w = bboxes[t*4+2], h = bboxes[t*4+3];
    float cx = x + 0.5f*w, cy = y + 0.5f*h;

    int best = 0; float bd = 3.4e38f;
    #pragma unroll
    for (int j = 0; j < 9; ++j) {
        float d = fabsf(w - anchors[2*j]) + fabsf(h - anchors[2*j+1]);
        if (d < bd) { bd = d; best = j; }   // first-min wins (argmin semantics)
    }
    int s = best / APS, a = best % APS;
    float tw = logf(w / anchors[2*best+0]);
    float th = logf(h / anchors[2*best+1]);

    int G   = (s == 0) ? G0s : ((s == 1) ? G1s : G2s);
    int off = (s == 0) ? 0   : ((s == 1) ? R0  : (R0 + R1));

    float fx = cx * (float)G, fy = cy * (float)G;
    int ix = (int)floorf(fx), iy = (int)floorf(fy);
    const float eps = 1e-8f;
    float gx = (fx - (float)ix) + eps;
    float gy = (fy - (float)iy) + eps;
    float tx = -logf(1.0f/gx - 1.0f);
    float ty = -logf(1.0f/gy - 1.0f);

    int b = t / NBOX;
    int row = off + (((b*G + iy)*G + ix)*APS + a);

    tvals[t] = make_float4(tx, ty, tw, th);
    labs[t]  = labels[t];
    atomicMax(&winner[row], t);
}

// ---------------- Kernel 2: TDM-staged conf + fused loss reduction --------------
__global__ void __launch_bounds__(256)
k_main(const float* __restrict__ p0,
       const float* __restrict__ p1,
       const float* __restrict__ p2,
       const int*    __restrict__ winner,
       const float4* __restrict__ tvals,
       const int*    __restrict__ labs,
       float* __restrict__ acc) {
    __shared__ float confbuf[TILE];

    int bid = blockIdx.x;
    const float* base; int rows_total, chunk, scale_off;
    if (bid < NBLK0)              { base = p0; rows_total = R0; chunk = bid;                 scale_off = 0;       }
    else if (bid < NBLK0 + NBLK1) { base = p1; rows_total = R1; chunk = bid - NBLK0;         scale_off = R0;      }
    else                          { base = p2; rows_total = R2; chunk = bid - NBLK0 - NBLK1; scale_off = R0 + R1; }
    int start = chunk * TILE;
    int valid = rows_total - start; if (valid > TILE) valid = TILE;

#ifdef HAVE_TDM
    if (threadIdx.x < 32u) {   // wave 0 issues the DMA (TDM is wave-level, EXEC-agnostic)
        tdm_gather_conf(base + (size_t)start * ROWLEN + 4,
                        (unsigned)(uintptr_t)confbuf,         // low 32 bits = LDS offset
                        (unsigned)(rows_total - start));      // OOB rows zero-filled
    }
    __syncthreads();
#else
    for (int i = threadIdx.x; i < TILE; i += blockDim.x) {
        int pi = i + (int)blockDim.x;
        if (start + pi < rows_total && pi < TILE)
            __builtin_prefetch(base + (size_t)(start + pi) * ROWLEN + 4, 0, 1);
        confbuf[i] = (i < valid) ? base[(size_t)(start + i) * ROWLEN + 4] : 0.0f;
    }
    __syncthreads();
#endif

    float confP = 0.f, coordP = 0.f, classP = 0.f, nobjP = 0.f;

    for (int k = threadIdx.x; k < TILE; k += blockDim.x) {
        if (k < valid) {
            int r = scale_off + start + k;
            int rp = r + (int)blockDim.x;
            __builtin_prefetch(&winner[(rp < TOTROWS) ? rp : r], 0, 1);

            int w = winner[r];
            float z = (w >= 0) ? 1.0f : 0.0f;
            confP += bce_logits(confbuf[k], z);

            if (w >= 0) {
                nobjP += 1.0f;
                const float* row = base + (size_t)(start + k) * ROWLEN;
                float4 tv = tvals[w];
                int lab = labs[w];
                float d0 = row[0]-tv.x, d1 = row[1]-tv.y;
                float d2 = row[2]-tv.z, d3 = row[3]-tv.w;
                coordP += d0*d0 + d1*d1 + d2*d2 + d3*d3;
                float cp = 0.f;
                #pragma unroll 4
                for (int cc = 0; cc < NUM_CLASSES; ++cc) {
                    float zz = (cc == lab) ? 1.0f : 0.0f;
                    cp += bce_logits(row[5+cc], zz);
                }
                classP += cp;
            }
        }
    }

    // Uniform end-of-kernel path: EXEC all-1s -> WMMA legal.
    wave_wmma_reduce_atomic(confP,  acc + 0);
    wave_wmma_reduce_atomic(coordP, acc + 1);
    wave_wmma_reduce_atomic(classP, acc + 2);
    wave_wmma_reduce_atomic(nobjP,  acc + 3);
}

// ---------------- Kernel 3: finalize scalars ------------------------------------
__global__ void k_final(const float* __restrict__ acc, float* __restrict__ out) {
    if (blockIdx.x == 0 && threadIdx.x == 0) {
        float conf_s  = acc[0];
        float coord_s = acc[1];
        float class_s = acc[2];
        float nobj    = fmaxf(acc[3], 1.0f);
        float coord_loss = 0.05f * coord_s / (nobj * 4.0f);
        float conf_loss  = 1.0f  * conf_s  / (float)TOTROWS;
        float class_loss = 0.5f  * class_s / (nobj * (float)NUM_CLASSES);
        out[0] = coord_loss + conf_loss + class_loss;
        out[1] = coord_loss;
        out[2] = conf_loss;
        out[3] = class_loss;
    }
}

extern "C" void kernel_launch(void* const* d_in, const int* in_sizes, int n_in,
                              void* d_out, int out_size, void* d_ws, size_t ws_size,
                              hipStream_t stream) {
    const float* p0      = (const float*)d_in[0];
    const float* p1      = (const float*)d_in[1];
    const float* p2      = (const float*)d_in[2];
    const float* bboxes  = (const float*)d_in[3];
    const int*   labels  = (const int*)  d_in[4];
    const float* anchors = (const float*)d_in[5];
    float* out = (float*)d_out;

    // workspace layout (d_ws 256B aligned):
    //  [0,16)              : 4 float accumulators (conf, coord, class, nobj)
    //  [16, 16+TOT*4)      : int winner[TOTROWS]
    //  then NB float4 tvals (16 + 681408*4 = 2725648, 16B aligned)
    //  then NB int labels
    char* ws = (char*)d_ws;
    float*  acc    = (float*)ws;
    int*    winner = (int*)(ws + 16);
    float4* tvals  = (float4*)(ws + 16 + (size_t)TOTROWS * 4);
    int*    labs   = (int*)(ws + 16 + (size_t)TOTROWS * 4 + (size_t)NB * 16);

    k_init  <<<1024, 256, 0, stream>>>(winner, acc);
    k_assign<<<(NB + 255)/256, 256, 0, stream>>>(bboxes, labels, anchors, winner, tvals, labs);
    k_main  <<<NBLK, 256, 0, stream>>>(p0, p1, p2, winner, tvals, labs, acc);
    k_final <<<1, 64, 0, stream>>>(acc, out);
}